// LambdaLayer_6330781794708
// MI455X (gfx1250) — compile-verified
//
#include <hip/hip_runtime.h>
#include <hip/hip_bf16.h>
#include <math.h>

// ---------------- problem constants ----------------
#define BB      32
#define NN      4096
#define DIN     512
#define HEADS   4
#define UU      4
#define KD      5
#define KS      7
#define VV      128
#define MROWS   (BB * NN)        // 131072
#define NCAT    768              // 512 (wv) + 20 (wq) + 20 (wk) + 216 pad

typedef float v2f __attribute__((ext_vector_type(2)));
typedef float v8f __attribute__((ext_vector_type(8)));

// ---------------- workspace layout (bytes) ----------------
// v_buf : MROWS*512 f32                = 268435456
// q_buf : MROWS*20  f32                =  10485760
// k_buf : B*20*N    f32 (transposed)   =  10485760
// lc    : B*5*128   f32                =     81920
// wcat  : 512*768   f32                =   1572864
#define WS_VBUF   0
#define WS_QBUF   (WS_VBUF + (size_t)MROWS * 512 * 4)
#define WS_KBUF   (WS_QBUF + (size_t)MROWS * 20 * 4)
#define WS_LC     (WS_KBUF + (size_t)BB * 20 * NN * 4)
#define WS_WCAT   (WS_LC   + (size_t)BB * KD * VV * 4)

// ---------------- kernel 0: pack [wv | wq | wk | 0] ----------------
__global__ __launch_bounds__(256) void pack_w_kernel(
    const float* __restrict__ wq, const float* __restrict__ wk,
    const float* __restrict__ wv, float* __restrict__ wcat) {
  int r = blockIdx.x;            // 0..511  (K row)
  for (int c = threadIdx.x; c < NCAT; c += 256) {
    float val = 0.0f;
    if (c < 512)       val = wv[(size_t)r * 512 + c];
    else if (c < 532)  val = wq[(size_t)r * 20 + (c - 512)];
    else if (c < 552)  val = wk[(size_t)r * 20 + (c - 532)];
    wcat[(size_t)r * NCAT + c] = val;
  }
}

// ---------------- kernel 0b: zero lc ----------------
__global__ __launch_bounds__(256) void zero_lc_kernel(float* __restrict__ lc) {
  int i = blockIdx.x * 256 + threadIdx.x;
  if (i < BB * KD * VV) lc[i] = 0.0f;
}

// ---------------- kernel 1: fused QKV GEMM via V_WMMA_F32_16X16X4_F32 --------
// C[M=131072, N=768] = x[M,512] @ wcat[512,768]
// block tile 128x256, 8 waves (2x4), each wave 64x64 (4x4 WMMA tiles)
#define LDSTR 34   // float stride for [row][k] LDS tiles (k-chunk = 32, +2 pad)

__global__ __launch_bounds__(256) void gemm_qkv_kernel(
    const float* __restrict__ x, const float* __restrict__ wcat,
    float* __restrict__ v_buf, float* __restrict__ q_buf,
    float* __restrict__ k_buf) {
  __shared__ float As[128 * LDSTR];   // [m][k]
  __shared__ float Bs[256 * LDSTR];   // [n][k] (transposed on load)

  const int m0   = blockIdx.x * 128;
  const int n0   = blockIdx.y * 256;
  const int t    = threadIdx.x;
  const int lane = t & 31;
  const int wave = t >> 5;      // 0..7
  const int wm   = wave & 1;    // 0..1 : which 64-row half
  const int wn   = wave >> 1;   // 0..3 : which 64-col quarter
  const int l16  = lane & 15;
  const int lhi  = lane >> 4;   // 0 or 1

  v8f acc[4][4];
  const v8f vzero = {0.f, 0.f, 0.f, 0.f, 0.f, 0.f, 0.f, 0.f};
#pragma unroll
  for (int i = 0; i < 4; ++i)
#pragma unroll
    for (int j = 0; j < 4; ++j) acc[i][j] = vzero;

  for (int kc = 0; kc < DIN; kc += 32) {
    // stage A: 128 rows x 32 k (4096 floats, 16 per thread)
#pragma unroll
    for (int i = 0; i < 16; ++i) {
      int idx = t + i * 256;          // 0..4095
      int r = idx >> 5;               // m within tile
      int c = idx & 31;               // k within chunk
      As[r * LDSTR + c] = x[(size_t)(m0 + r) * DIN + kc + c];
    }
    // stage B transposed: 32 k x 256 n -> Bs[n][k] (8192 floats, 32 per thread)
#pragma unroll
    for (int i = 0; i < 32; ++i) {
      int idx = t + i * 256;          // 0..8191
      int r = idx >> 8;               // k within chunk
      int c = idx & 255;              // n within tile
      Bs[c * LDSTR + r] = wcat[(size_t)(kc + r) * NCAT + n0 + c];
    }
    __syncthreads();

#pragma unroll
    for (int kk = 0; kk < 32; kk += 4) {
      const int kb = kk + 2 * lhi;    // lanes 0-15 hold K,K+1 ; 16-31 hold K+2,K+3
      v2f afrag[4], bfrag[4];
#pragma unroll
      for (int i = 0; i < 4; ++i)
        afrag[i] = *(const v2f*)&As[(wm * 64 + i * 16 + l16) * LDSTR + kb];
#pragma unroll
      for (int j = 0; j < 4; ++j)
        bfrag[j] = *(const v2f*)&Bs[(wn * 64 + j * 16 + l16) * LDSTR + kb];
#pragma unroll
      for (int i = 0; i < 4; ++i)
#pragma unroll
        for (int j = 0; j < 4; ++j)
          acc[i][j] = __builtin_amdgcn_wmma_f32_16x16x4_f32(
              false, afrag[i], false, bfrag[j], (short)0, acc[i][j],
              false, false);
    }
    __syncthreads();
  }

  // epilogue: route columns -> v / q / k(transposed) / discard
#pragma unroll
  for (int i = 0; i < 4; ++i) {
#pragma unroll
    for (int j = 0; j < 4; ++j) {
      const int col = n0 + wn * 64 + j * 16 + l16;
#pragma unroll
      for (int r = 0; r < 8; ++r) {
        const int row = m0 + wm * 64 + i * 16 + r + 8 * lhi;
        const float val = acc[i][j][r];
        if (col < 512) {
          v_buf[(size_t)row * 512 + col] = val;
        } else if (col < 532) {
          q_buf[(size_t)row * 20 + (col - 512)] = val;
        } else if (col < 552) {
          const int b = row >> 12;        // N == 4096
          const int n = row & (NN - 1);
          k_buf[((size_t)(b * 20 + (col - 532))) * NN + n] = val;
        }
      }
    }
  }
}

// ---------------- kernel 2: softmax over n (in place on k_buf) --------------
__global__ __launch_bounds__(256) void softmax_k_kernel(float* __restrict__ k_buf) {
  __shared__ float red[256];
  float* p = k_buf + (size_t)blockIdx.x * NN;   // one (b, u*5+kd) row
  const int t = threadIdx.x;

  float vals[16];
  float mx = -INFINITY;
#pragma unroll
  for (int i = 0; i < 16; ++i) {
    vals[i] = p[t + i * 256];
    mx = fmaxf(mx, vals[i]);
  }
  red[t] = mx; __syncthreads();
  for (int s = 128; s > 0; s >>= 1) {
    if (t < s) red[t] = fmaxf(red[t], red[t + s]);
    __syncthreads();
  }
  mx = red[0]; __syncthreads();

  float sum = 0.0f;
#pragma unroll
  for (int i = 0; i < 16; ++i) {
    vals[i] = __expf(vals[i] - mx);
    sum += vals[i];
  }
  red[t] = sum; __syncthreads();
  for (int s = 128; s > 0; s >>= 1) {
    if (t < s) red[t] += red[t + s];
    __syncthreads();
  }
  const float inv = 1.0f / red[0];
#pragma unroll
  for (int i = 0; i < 16; ++i) p[t + i * 256] = vals[i] * inv;
}

// ---------------- kernel 3: lc[b,kd,vv] += sum_{u,n} softk * v ---------------
__global__ __launch_bounds__(256) void lc_accum_kernel(
    const float* __restrict__ v_buf, const float* __restrict__ softk,
    float* __restrict__ lc) {
  const int b  = blockIdx.x;
  const int n0 = blockIdx.y * 128;
  const int t  = threadIdx.x;
  const int vv = t & 127;
  const int up = t >> 7;          // u pair: {0,1} or {2,3}

  float acc[KD] = {0.f, 0.f, 0.f, 0.f, 0.f};
  for (int nn = 0; nn < 128; ++nn) {
    const int n = n0 + nn;
    const size_t vrow = ((size_t)b * NN + n) * 512;
#pragma unroll
    for (int uu = 0; uu < 2; ++uu) {
      const int u = up * 2 + uu;
      const float vval = v_buf[vrow + u * VV + vv];
      const float* skp = softk + ((size_t)(b * 20 + u * KD)) * NN + n;
#pragma unroll
      for (int kd = 0; kd < KD; ++kd) acc[kd] += skp[(size_t)kd * NN] * vval;
    }
  }
#pragma unroll
  for (int kd = 0; kd < KD; ++kd)
    atomicAdd(&lc[(b * KD + kd) * VV + vv], acc[kd]);
}

// ---------------- kernel 4: position conv + y = q·(lc + lp) -----------------
__global__ __launch_bounds__(256) void final_kernel(
    const float* __restrict__ v_buf, const float* __restrict__ q_buf,
    const float* __restrict__ lc, const float* __restrict__ pos_w,
    const float* __restrict__ pos_b, float* __restrict__ out) {
  __shared__ float lcs[KD * VV];          // 640
  __shared__ float pws[KS * UU * KD];     // 140
  __shared__ float pbs[KD];

  const int b = blockIdx.x;
  const int t = threadIdx.x;
  for (int i = t; i < KD * VV; i += 256) lcs[i] = lc[b * KD * VV + i];
  if (t < KS * UU * KD) pws[t] = pos_w[t];
  if (t < KD) pbs[t] = pos_b[t];
  __syncthreads();

  const int n  = blockIdx.y * 2 + (t >> 7);   // 2 sequence positions per block
  const int vv = t & 127;

  float lp[KD];
#pragma unroll
  for (int kd = 0; kd < KD; ++kd) lp[kd] = pbs[kd];

#pragma unroll
  for (int j = 0; j < KS; ++j) {
    const int ns = n + j - (KS / 2);
    if (ns < 0 || ns >= NN) continue;     // SAME padding -> zero
    const size_t vrow = ((size_t)b * NN + ns) * 512;
#pragma unroll
    for (int u = 0; u < UU; ++u) {
      const float vval = v_buf[vrow + u * VV + vv];
      const float* pw = &pws[(j * UU + u) * KD];
#pragma unroll
      for (int kd = 0; kd < KD; ++kd) lp[kd] += vval * pw[kd];
    }
  }

  float lam[KD];
#pragma unroll
  for (int kd = 0; kd < KD; ++kd) lam[kd] = lp[kd] + lcs[kd * VV + vv];

  const float* qp  = q_buf + ((size_t)b * NN + n) * 20;
  const size_t orow = ((size_t)b * NN + n) * 512;
#pragma unroll
  for (int h = 0; h < HEADS; ++h) {
    float y = 0.0f;
#pragma unroll
    for (int kd = 0; kd < KD; ++kd) y += qp[h * KD + kd] * lam[kd];
    out[orow + h * VV + vv] = y;
  }
}

// ---------------- launcher ----------------
extern "C" void kernel_launch(void* const* d_in, const int* in_sizes, int n_in,
                              void* d_out, int out_size, void* d_ws, size_t ws_size,
                              hipStream_t stream) {
  const float* x     = (const float*)d_in[0];
  const float* wq    = (const float*)d_in[1];
  const float* wk    = (const float*)d_in[2];
  const float* wv    = (const float*)d_in[3];
  const float* pos_w = (const float*)d_in[4];
  const float* pos_b = (const float*)d_in[5];
  float* out = (float*)d_out;

  char* ws = (char*)d_ws;
  float* v_buf = (float*)(ws + WS_VBUF);
  float* q_buf = (float*)(ws + WS_QBUF);
  float* k_buf = (float*)(ws + WS_KBUF);
  float* lc    = (float*)(ws + WS_LC);
  float* wcat  = (float*)(ws + WS_WCAT);

  pack_w_kernel<<<DIN, 256, 0, stream>>>(wq, wk, wv, wcat);
  zero_lc_kernel<<<(BB * KD * VV + 255) / 256, 256, 0, stream>>>(lc);
  gemm_qkv_kernel<<<dim3(MROWS / 128, NCAT / 256), 256, 0, stream>>>(
      x, wcat, v_buf, q_buf, k_buf);
  softmax_k_kernel<<<BB * 20, 256, 0, stream>>>(k_buf);
  lc_accum_kernel<<<dim3(BB, NN / 128), 256, 0, stream>>>(v_buf, k_buf, lc);
  final_kernel<<<dim3(BB, NN / 2), 256, 0, stream>>>(v_buf, q_buf, lc, pos_w,
                                                     pos_b, out);
}